// EdgeLoss_40381282517094
// MI455X (gfx1250) — compile-verified
//
#include <hip/hip_runtime.h>
#include <math.h>

// ---------------------------------------------------------------------------
// Fused Canny edge loss for MI455X (gfx1250, wave32).
//  - one workgroup per 64x64 tile, 92x92 gray region in LDS (halo 14 covers
//    blur(2)+sobel(1)+nms(1)+10 hysteresis iterations)
//  - separable 5-tap Gaussian implemented as banded-matrix products using
//    V_WMMA_F32_16X16X4_F32 (5 K=4 chunks per 16x16 tile, both passes share
//    one set of per-lane coefficient fragments by layout symmetry)
//  - destination LDS buffers carry slop rows so the WMMA D-store needs only
//    a single per-lane column guard (no per-VGPR EXEC juggling)
//  - both images processed in the same block; |e1-e2| reduced locally,
//    single global_atomic_add_f32 per block; tiny finalize kernel scales.
// ---------------------------------------------------------------------------

typedef __attribute__((ext_vector_type(2))) float v2f;
typedef __attribute__((ext_vector_type(8))) float v8f;

#define TS     64
#define HALO   14
#define RW     92            // TS + 2*HALO (valid data rows/cols)
#define RS     93            // LDS row stride (odd -> bank friendly)
#define AROWS  98            // bufA rows: vertical pass writes rows up to 97
#define BROWS  96            // bufB rows: horizontal pass writes rows up to 95
#define NPIXA  (AROWS*RS)    // 9114 floats
#define NPIXB  (BROWS*RS)    // 8928 floats
#define NPIX   (RW*RS)       // 8556 (byte-map arrays)
#define NT     256
#define NWAVES 8
#define IMG    512

__device__ __forceinline__ int clampi(int v, int lo, int hi) {
  return v < lo ? lo : (v > hi ? hi : v);
}

__global__ void zero_acc_kernel(float* acc) { acc[0] = 0.0f; }

__global__ void finalize_kernel(const float* acc, float* out) {
  out[0] = acc[0] * (1.0f / (16.0f * 512.0f * 512.0f));
}

__global__ __launch_bounds__(NT)
void canny_diff_kernel(const float* __restrict__ yhat,
                       const float* __restrict__ yref,
                       float* __restrict__ acc)
{
  extern __shared__ unsigned char smem_raw[];
  float*         bufA  = (float*)smem_raw;          // gray, later vblur
  float*         bufB  = bufA + NPIXA;              // hblur, later magnitude
  float*         red   = bufB + NPIXB;              // NT-wide reduction
  unsigned char* dirb  = (unsigned char*)(red + NT);
  unsigned char* stA   = dirb + NPIX;
  unsigned char* stB   = stA + NPIX;
  unsigned char* edge1 = stB + NPIX;                // 64*64

  const int tid  = threadIdx.x;
  const int lane = tid & 31;
  const int wave = tid >> 5;
  const int l16  = lane & 15;
  const int half = lane >> 4;       // 0 or 1
  const int kb   = half << 1;       // base K index of this lane's pair {kb,kb+1}

  // 5-tap Gaussian, sigma=1 (kornia default)
  const float g5[5] = {0.05448868f, 0.24420136f, 0.40261996f,
                       0.24420136f, 0.05448868f};

  // Banded convolution matrix fragments. For chunk j the coefficient value at
  // (row r=l16 / col K=kb+c) is g[4j+kb+c - l16] when in band, else 0.
  // By the A/B layout symmetry this one fragment set serves as B for the
  // horizontal pass and as A for the vertical pass.
  v2f wcoef[5];
#pragma unroll
  for (int j = 0; j < 5; ++j) {
    int d0 = 4 * j + kb - l16;
    float x = (d0     >= 0 && d0     < 5) ? g5[d0]     : 0.0f;
    float y = (d0 + 1 >= 0 && d0 + 1 < 5) ? g5[d0 + 1] : 0.0f;
    wcoef[j].x = x;
    wcoef[j].y = y;
  }

  const int b  = blockIdx.z;
  const int ox = blockIdx.x * TS - HALO;
  const int oy = blockIdx.y * TS - HALO;

  const int offy[8] = {0, 1, 1, 1, 0, -1, -1, -1};
  const int offx[8] = {1, 1, 0, -1, -1, -1, 0, 1};

  float diffs = 0.0f;

  for (int pass = 0; pass < 2; ++pass) {
    const float* img   = pass ? yref : yhat;
    const float* plane = img + (size_t)b * 3 * IMG * IMG;

    // ---- stage 0: grayscale into LDS (clamp-to-edge halo) ----
    for (int i = tid; i < RW * RW; i += NT) {
      int ly = i / RW, lx = i - ly * RW;
      int py = clampi(oy + ly, 0, IMG - 1);
      int px = clampi(ox + lx, 0, IMG - 1);
      const float* p0 = plane + (size_t)py * IMG + px;
      bufA[ly * RS + lx] =
          0.299f * p0[0] + 0.587f * p0[IMG * IMG] + 0.114f * p0[2 * IMG * IMG];
    }
    __syncthreads();

    // ---- stage 1: horizontal Gaussian = A(data 16x4) x B(band 4x16), WMMA ----
    // output cols 2..89 (6x16 col tiles), rows 0..91 (6x16 row tiles, slop to 95)
    for (int t = wave; t < 36; t += NWAVES) {
      int R0 = 16 * (t / 6);
      int C0 = 2 + 16 * (t % 6);
      v8f d = {};
      const float* arow = bufA + clampi(R0 + l16, 0, RW - 1) * RS;
#pragma unroll
      for (int j = 0; j < 5; ++j) {
        int c0 = C0 - 2 + 4 * j + kb;
        v2f a;
        a.x = arow[clampi(c0,     0, RW - 1)];
        a.y = arow[clampi(c0 + 1, 0, RW - 1)];
        d = __builtin_amdgcn_wmma_f32_16x16x4_f32(
            false, a, false, wcoef[j], (short)0, d, false, false);
      }
      int col = C0 + l16;
      if (col < 90) {                       // single per-lane guard
        float* p = bufB + (R0 + (half << 3)) * RS + col;
#pragma unroll
        for (int v = 0; v < 8; ++v) p[v * RS] = d[v];   // rows <= 95 < BROWS
      }
    }
    __syncthreads();

    // ---- stage 2: vertical Gaussian = A(band 16x4) x B(data 4x16), WMMA ----
    // output region rows/cols 2..89 (row slop to 97)
    for (int t = wave; t < 36; t += NWAVES) {
      int R0 = 2 + 16 * (t / 6);
      int C0 = 2 + 16 * (t % 6);
      v8f d = {};
      int col = clampi(C0 + l16, 0, RW - 1);
#pragma unroll
      for (int j = 0; j < 5; ++j) {
        int r0 = R0 - 2 + 4 * j + kb;
        v2f bfrag;
        bfrag.x = bufB[clampi(r0,     0, RW - 1) * RS + col];
        bfrag.y = bufB[clampi(r0 + 1, 0, RW - 1) * RS + col];
        d = __builtin_amdgcn_wmma_f32_16x16x4_f32(
            false, wcoef[j], false, bfrag, (short)0, d, false, false);
      }
      int c = C0 + l16;
      if (c < 90) {                         // single per-lane guard
        float* p = bufA + (R0 + (half << 3)) * RS + c;
#pragma unroll
        for (int v = 0; v < 8; ++v) p[v * RS] = d[v];   // rows <= 97 < AROWS
      }
    }
    __syncthreads();

    // ---- stage 3: Sobel, magnitude, quantized direction on [3..88]^2 ----
    for (int i = tid; i < 86 * 86; i += NT) {
      int y = 3 + i / 86, x = 3 + i - (i / 86) * 86;
      const float* p = &bufA[y * RS + x];
      float gx = (p[-RS + 1] + 2.0f * p[1] + p[RS + 1])
               - (p[-RS - 1] + 2.0f * p[-1] + p[RS - 1]);
      float gy = (p[RS - 1] + 2.0f * p[RS] + p[RS + 1])
               - (p[-RS - 1] + 2.0f * p[-RS] + p[-RS + 1]);
      float mag = sqrtf(gx * gx + gy * gy + 1e-6f);
      float a45 = atan2f(gy, gx) * 1.27323954f;   // radians -> units of 45 deg
      int k = ((int)rintf(a45) + 8) & 7;
      bufB[y * RS + x] = mag;     // hblur fully consumed; reuse as magnitude
      dirb[y * RS + x] = (unsigned char)k;
    }
    __syncthreads();

    // ---- stage 4: NMS + double threshold on [4..87]^2 ----
    for (int i = tid; i < NPIX; i += NT) { stA[i] = 0; stB[i] = 0; }
    __syncthreads();
    for (int i = tid; i < 84 * 84; i += NT) {
      int y = 4 + i / 84, x = 4 + i - (i / 84) * 84;
      int p = y * RS + x;
      float m = bufB[p];
      int k = dirb[p];
      float pd = m - bufB[(y + offy[k]) * RS + (x + offx[k])];
      float nd = m - bufB[(y - offy[k]) * RS + (x - offx[k])];
      float mm = (fminf(pd, nd) > 0.0f) ? m : 0.0f;
      stA[p] = (mm > 0.2f) ? 2 : ((mm > 0.1f) ? 1 : 0);
    }
    __syncthreads();

    // ---- stage 5: hysteresis, 10 ping-pong dilation iterations ----
    unsigned char* src = stA;
    unsigned char* dst = stB;
    for (int it = 0; it < 10; ++it) {
      for (int i = tid; i < 84 * 84; i += NT) {
        int y = 4 + i / 84, x = 4 + i - (i / 84) * 84;
        int p = y * RS + x;
        unsigned char s = src[p];
        if (s == 1) {
          int any = 0;
#pragma unroll
          for (int dy = -1; dy <= 1; ++dy)
#pragma unroll
            for (int dx = -1; dx <= 1; ++dx)
              any |= (src[p + dy * RS + dx] == 2);
          if (any) s = 2;
        }
        dst[p] = s;
      }
      __syncthreads();
      unsigned char* tmp = src; src = dst; dst = tmp;
    }

    // ---- stage 6: central 64x64 edge map / diff ----
    for (int i = tid; i < 64 * 64; i += NT) {
      int y = HALO + (i >> 6), x = HALO + (i & 63);
      unsigned char e = (src[y * RS + x] == 2) ? 1 : 0;
      if (pass == 0) edge1[i] = e;
      else           diffs += fabsf((float)edge1[i] - (float)e);
    }
    __syncthreads();
  }

  // ---- block reduction + one global float atomic ----
  red[tid] = diffs;
  __syncthreads();
  for (int s = NT / 2; s > 0; s >>= 1) {
    if (tid < s) red[tid] += red[tid + s];
    __syncthreads();
  }
  if (tid == 0) atomicAdd(acc, red[0]);
}

extern "C" void kernel_launch(void* const* d_in, const int* in_sizes, int n_in,
                              void* d_out, int out_size, void* d_ws, size_t ws_size,
                              hipStream_t stream) {
  const float* yhat = (const float*)d_in[0];
  const float* yref = (const float*)d_in[1];
  float* out = (float*)d_out;
  float* accp = (float*)d_ws;

  zero_acc_kernel<<<1, 1, 0, stream>>>(accp);

  dim3 grid(IMG / TS, IMG / TS, 16);   // 8 x 8 x 16 tiles
  size_t smem = (size_t)(NPIXA + NPIXB + NT) * sizeof(float)  // bufA,bufB,red
              + (size_t)NPIX * 3                               // dirb,stA,stB
              + 64 * 64;                                       // edge1
  smem = (smem + 15) & ~(size_t)15;
  canny_diff_kernel<<<grid, NT, smem, stream>>>(yhat, yref, accp);

  finalize_kernel<<<1, 1, 0, stream>>>(accp, out);
}